// PureQIGKernelTemplate_40965398069672
// MI455X (gfx1250) — compile-verified
//
#include <hip/hip_runtime.h>
#include <math.h>

#define EPS 1e-8f
#define TT 512
#define DD 64
#define NB 4

typedef float v2f __attribute__((ext_vector_type(2)));
typedef float v8f __attribute__((ext_vector_type(8)));

__device__ __forceinline__ float softplus_f(float v) {
  // numerically stable softplus
  return fmaxf(v, 0.f) + log1pf(__expf(-fabsf(v)));
}

__device__ __forceinline__ float hred16(float v) {
  // sum across the 16-lane half of a wave32 (masks 1..8 stay within a half)
  v += __shfl_xor(v, 1, 32);
  v += __shfl_xor(v, 2, 32);
  v += __shfl_xor(v, 4, 32);
  v += __shfl_xor(v, 8, 32);
  return v;
}
__device__ __forceinline__ float wred32(float v) {
  v = hred16(v);
  v += __shfl_xor(v, 16, 32);
  return v;
}

// ---------------------------------------------------------------------------
// Kernel 1: p = simplex projection of softplus(x); emit s = sqrt(p), r = 1/sqrt(p)
// One wave per row of 64; lane handles elements [lane] and [lane+32].
// ---------------------------------------------------------------------------
__global__ __launch_bounds__(256) void k_simplex_sqrt(const float* __restrict__ x,
                                                      float* __restrict__ s,
                                                      float* __restrict__ r) {
  const int row  = blockIdx.x * 8 + (threadIdx.x >> 5);
  const int lane = threadIdx.x & 31;
  const float* xr = x + row * DD;
  float a = softplus_f(xr[lane]);
  float b = softplus_f(xr[lane + 32]);
  float sum  = wred32(a + b);
  float inv  = 1.f / (sum + EPS);
  float p0 = fmaxf(a * inv, EPS);
  float p1 = fmaxf(b * inv, EPS);
  float sum2 = wred32(p0 + p1);
  float inv2 = 1.f / (sum2 + EPS);
  float q0 = sqrtf(p0 * inv2);
  float q1 = sqrtf(p1 * inv2);
  s[row * DD + lane]      = q0;
  s[row * DD + lane + 32] = q1;
  r[row * DD + lane]      = 1.f / q0;
  r[row * DD + lane + 32] = 1.f / q1;
}

// ---------------------------------------------------------------------------
// Kernel 2: fused Fisher-Rao attention + residual blend for one 16-row i-tile.
// grid = B * (T/16) = 128 blocks, 4 wave32s each; wave w handles j-tiles
// w, w+4, ... Per j-tile:
//   inner = S_i S_j^T + (eps/2) R_i R_j^T       (32x v_wmma_f32_16x16x4_f32)
//   e     = exp(-2*acos(clip(inner)))           (VALU transcendentals)
//   accN += W_tile(16x16) @ X_tile(16x64)       (16x v_wmma_f32_16x16x4_f32)
// Wave partials (accN, rowsum) are combined in LDS, then
//   x' = 0.5(1-c)*x + 0.5(1+c)*(acc / rowsum).
// ---------------------------------------------------------------------------
__global__ __launch_bounds__(128) void k_attn_fused(const float* __restrict__ s,
                                                    const float* __restrict__ rq,
                                                    const float* __restrict__ xin,
                                                    float* __restrict__ xout,
                                                    const float* __restrict__ rs) {
  const int wave = threadIdx.x >> 5;
  const int lane = threadIdx.x & 31;
  const int half = lane >> 4;
  const int l16  = lane & 15;
  const int tileId = blockIdx.x;            // 0..127
  const int b  = tileId >> 5;               // 32 i-tiles per batch
  const int i0 = (tileId & 31) << 4;
  const float* Sb = s   + b * TT * DD;
  const float* Rb = rq  + b * TT * DD;
  const float* Xb = xin + b * TT * DD;
  float* Ob = xout + b * TT * DD;

  __shared__ float wt[4][16][17];           // per-wave weight tile (padded)
  __shared__ float pacc[4][NB][16][16];     // per-wave x_attn partials
  __shared__ float prs[4][16];              // per-wave rowsum partials
  __shared__ float frs[16];

  const float c     = 0.01f * rs[0];
  const float alpha = 0.5f * (1.f - c);
  const float beta  = 0.5f * (1.f + c);
  const v8f zero8 = {0.f, 0.f, 0.f, 0.f, 0.f, 0.f, 0.f, 0.f};

  v8f accN[NB];
#pragma unroll
  for (int nb = 0; nb < NB; ++nb) accN[nb] = zero8;
  float rowsum[8];
#pragma unroll
  for (int q = 0; q < 8; ++q) rowsum[q] = 0.f;

  // A-fragment base offset for this wave's 16 i-rows (v_wmma f32 16x16x4 layout:
  // lanes 0-15 carry M rows with K={0,1}; lanes 16-31 carry K={2,3})
  const int aOff = (i0 + l16) * DD + 2 * half;

  for (int jt = wave; jt < 32; jt += 4) {
    const int j0  = jt << 4;
    const int bOff = (j0 + l16) * DD + 2 * half;

    // ---- inner product GEMMs over K = D = 64 ----
    v8f c1 = zero8, c2 = zero8;
#pragma unroll
    for (int kk = 0; kk < 16; ++kk) {
      v2f As = *(const v2f*)(Sb + aOff + 4 * kk);
      v2f Bs = *(const v2f*)(Sb + bOff + 4 * kk);
      c1 = __builtin_amdgcn_wmma_f32_16x16x4_f32(false, As, false, Bs, (short)0, c1, false, false);
      v2f Ar = *(const v2f*)(Rb + aOff + 4 * kk);
      v2f Br = *(const v2f*)(Rb + bOff + 4 * kk);
      c2 = __builtin_amdgcn_wmma_f32_16x16x4_f32(false, Ar, false, Br, (short)0, c2, false, false);
    }

    // ---- unnormalized softmax weights; stage to LDS; row-sum partials ----
#pragma unroll
    for (int q = 0; q < 8; ++q) {
      float inner = c1[q] + (0.5f * EPS) * c2[q];   // ~= sum_d sqrt(p_i p_j + eps)
      inner = fminf(fmaxf(inner, -1.f + 1e-6f), 1.f - 1e-6f);
      float e = __expf(-2.f * acosf(inner));
      wt[wave][q + 8 * half][l16] = e;
      rowsum[q] += hred16(e);                        // same value across the half
    }

    // ---- accN += W(16x16) @ X(16x64), K = 16 (j within tile) ----
#pragma unroll
    for (int nb = 0; nb < NB; ++nb) {
#pragma unroll
      for (int kk = 0; kk < 4; ++kk) {
        v2f A;
        A.x = wt[wave][l16][4 * kk + 2 * half];
        A.y = wt[wave][l16][4 * kk + 2 * half + 1];
        const float* xp = Xb + (j0 + 4 * kk + 2 * half) * DD + nb * 16 + l16;
        v2f Bf;
        Bf.x = xp[0];
        Bf.y = xp[DD];
        accN[nb] = __builtin_amdgcn_wmma_f32_16x16x4_f32(false, A, false, Bf, (short)0, accN[nb], false, false);
      }
    }
  }

  // ---- stage wave partials ----
#pragma unroll
  for (int nb = 0; nb < NB; ++nb)
#pragma unroll
    for (int q = 0; q < 8; ++q)
      pacc[wave][nb][q + 8 * half][l16] = accN[nb][q];
  if (l16 == 0) {
#pragma unroll
    for (int q = 0; q < 8; ++q) prs[wave][q + 8 * half] = rowsum[q];
  }
  __syncthreads();

  if (threadIdx.x < 16)
    frs[threadIdx.x] = prs[0][threadIdx.x] + prs[1][threadIdx.x] +
                       prs[2][threadIdx.x] + prs[3][threadIdx.x];
  __syncthreads();

  // ---- combine + residual blend + writeout (16 x 64 tile, 8 elems/thread) ----
#pragma unroll
  for (int q = 0; q < 8; ++q) {
    int idx = threadIdx.x * 8 + q;          // 0..1023
    int nb  = idx >> 8;
    int rr  = (idx >> 4) & 15;
    int cc  = idx & 15;
    float acc = pacc[0][nb][rr][cc] + pacc[1][nb][rr][cc] +
                pacc[2][nb][rr][cc] + pacc[3][nb][rr][cc];
    float attn = acc / frs[rr];
    int row = i0 + rr, col = nb * 16 + cc;
    float xo = Xb[row * DD + col];
    Ob[row * DD + col] = alpha * xo + beta * attn;
  }
}

// ---------------------------------------------------------------------------
// Kernel 3a: per-(b,d) time mean
// ---------------------------------------------------------------------------
__global__ __launch_bounds__(64) void k_mean(const float* __restrict__ x,
                                             float* __restrict__ mu) {
  const int b = blockIdx.x, d = threadIdx.x;
  const float* xb = x + b * TT * DD;
  float sum = 0.f;
  for (int t = 0; t < TT; ++t) sum += xb[t * DD + d];
  mu[b * DD + d] = sum * (1.f / TT);
}

// ---------------------------------------------------------------------------
// Kernel 3b: cov = xc^T xc / (T+eps) via v_wmma_f32_16x16x4_f32, K = T = 512.
// grid = B blocks, 4 wave32s; wave handles 4 of the 16 (d,e) tiles.
// ---------------------------------------------------------------------------
__global__ __launch_bounds__(128) void k_cov(const float* __restrict__ x,
                                             const float* __restrict__ mu,
                                             float* __restrict__ cov) {
  const int wave = threadIdx.x >> 5, lane = threadIdx.x & 31;
  const int half = lane >> 4, l16 = lane & 15;
  const int b = blockIdx.x;
  const float* xb = x + b * TT * DD;
  const float* mb = mu + b * DD;
  const float invT = 1.f / ((float)TT + EPS);
  const v8f zero8 = {0.f, 0.f, 0.f, 0.f, 0.f, 0.f, 0.f, 0.f};

  for (int q = 0; q < 4; ++q) {
    const int tile = wave * 4 + q;
    const int d0 = (tile >> 2) << 4;
    const int e0 = (tile & 3) << 4;
    const float muA = mb[d0 + l16];
    const float muB = mb[e0 + l16];
    v8f cacc = zero8;
    for (int kk = 0; kk < 128; ++kk) {
      const int t0 = 4 * kk + 2 * half;
      v2f A, Bf;
      A.x  = xb[t0 * DD + d0 + l16] - muA;
      A.y  = xb[(t0 + 1) * DD + d0 + l16] - muA;
      Bf.x = xb[t0 * DD + e0 + l16] - muB;
      Bf.y = xb[(t0 + 1) * DD + e0 + l16] - muB;
      cacc = __builtin_amdgcn_wmma_f32_16x16x4_f32(false, A, false, Bf, (short)0, cacc, false, false);
    }
#pragma unroll
    for (int r8 = 0; r8 < 8; ++r8)
      cov[b * DD * DD + (d0 + r8 + 8 * half) * DD + e0 + l16] = cacc[r8] * invT;
  }
}

// ---------------------------------------------------------------------------
// Kernel 3c: phi = mean |offdiag(corr)|
// ---------------------------------------------------------------------------
__global__ __launch_bounds__(256) void k_phi(const float* __restrict__ cov,
                                             float* __restrict__ phi) {
  const int b = blockIdx.x;
  const float* cb = cov + b * DD * DD;
  __shared__ float sq[DD];
  __shared__ float red[256];
  if (threadIdx.x < DD)
    sq[threadIdx.x] = sqrtf(fmaxf(cb[threadIdx.x * DD + threadIdx.x], EPS));
  __syncthreads();
  float sum = 0.f;
  for (int idx = threadIdx.x; idx < DD * DD; idx += 256) {
    int d = idx >> 6, e = idx & 63;
    if (d != e) {
      float denom = fmaxf(sq[d] * sq[e], EPS);
      float corr  = fminf(fmaxf(cb[idx] / denom, -1.f), 1.f);
      sum += fabsf(corr);
    }
  }
  red[threadIdx.x] = sum;
  __syncthreads();
  for (int st = 128; st > 0; st >>= 1) {
    if (threadIdx.x < st) red[threadIdx.x] += red[threadIdx.x + st];
    __syncthreads();
  }
  if (threadIdx.x == 0) phi[b] = red[0] * (1.f / (DD * DD));
}

// ---------------------------------------------------------------------------
// Kernel 3d: kappa = mean_t ||x[b,t,:]||_2
// ---------------------------------------------------------------------------
__global__ __launch_bounds__(256) void k_kappa(const float* __restrict__ x,
                                               float* __restrict__ kappa) {
  const int b = blockIdx.x;
  const int wave = threadIdx.x >> 5, lane = threadIdx.x & 31;
  const float* xb = x + b * TT * DD;
  __shared__ float wsum[8];
  float acc = 0.f;
  for (int t = wave; t < TT; t += 8) {
    float v0 = xb[t * DD + lane];
    float v1 = xb[t * DD + lane + 32];
    float ss = wred32(v0 * v0 + v1 * v1);
    if (lane == 0) acc += sqrtf(ss);
  }
  if (lane == 0) wsum[wave] = acc;
  __syncthreads();
  if (threadIdx.x == 0) {
    float ssum = 0.f;
    for (int w = 0; w < 8; ++w) ssum += wsum[w];
    kappa[b] = ssum * (1.f / TT);
  }
}

// ---------------------------------------------------------------------------
extern "C" void kernel_launch(void* const* d_in, const int* in_sizes, int n_in,
                              void* d_out, int out_size, void* d_ws, size_t ws_size,
                              hipStream_t stream) {
  const float* x0 = (const float*)d_in[0];   // (4,512,64) f32
  const float* rs = (const float*)d_in[1];   // scalar f32
  float* out = (float*)d_out;                // [x 131072][phi 4][kappa 4]
  const int BTD = 4 * TT * DD;               // 131072

  float* ws    = (float*)d_ws;
  float* s_buf = ws;                         // BTD  (sqrt p)
  float* r_buf = ws + BTD;                   // BTD  (1/sqrt p)
  float* x1    = ws + 2 * BTD;               // BTD  (intermediate x)
  // s/r are dead after the 2nd attention pass -> overlap cov/mu into s_buf
  float* cov   = s_buf;                      // 4*64*64
  float* mu    = s_buf + 4 * DD * DD;        // 4*64

  float* xfin = out;                         // final x lands in d_out

  // iteration 0: d_in -> x1
  k_simplex_sqrt<<<256, 256, 0, stream>>>(x0, s_buf, r_buf);
  k_attn_fused<<<128, 128, 0, stream>>>(s_buf, r_buf, x0, x1, rs);
  // iteration 1: x1 -> d_out
  k_simplex_sqrt<<<256, 256, 0, stream>>>(x1, s_buf, r_buf);
  k_attn_fused<<<128, 128, 0, stream>>>(s_buf, r_buf, x1, xfin, rs);
  // statistics
  k_mean<<<4, 64, 0, stream>>>(xfin, mu);
  k_cov<<<4, 128, 0, stream>>>(xfin, mu, cov);
  k_phi<<<4, 256, 0, stream>>>(cov, out + BTD);
  k_kappa<<<4, 256, 0, stream>>>(xfin, out + BTD + 4);
}